// MTLoRALinear_13915694039699
// MI455X (gfx1250) — compile-verified
//
#include <hip/hip_runtime.h>

// MTLoRALinear fused kernel for MI455X (gfx1250, wave32, WMMA).
// out[0]   = x@W^T + b + (x@A_sh^T)@B_sh^T
// out[1+t] = x@W^T + b + scale_t * (x@A_t^T)@B_t^T
// Stage 1: rank-16 projections xa[5][M][16] into workspace (10 MB, L2-resident).
// Stage 2: LDS-tiled main GEMM + 5-way rank-16 expansion + fused 5-output store.
// Output writes use non-temporal hints (write-once 671 MB stream must not
// sweep the 192 MB L2 that keeps W/xa/x resident).

typedef float v2f __attribute__((ext_vector_type(2)));
typedef float v4f __attribute__((ext_vector_type(4)));
typedef float v8f __attribute__((ext_vector_type(8)));

#define B_DIM 8
#define S_DIM 4096
#define D_DIM 1024
#define O_DIM 1024
#define R_DIM 16
#define M_TOT (B_DIM * S_DIM)  // 32768 tokens

static __device__ __forceinline__ v8f wmma_f32(v2f a, v2f b, v8f c) {
  // V_WMMA_F32_16X16X4_F32: (neg_a, A, neg_b, B, c_mod, C, reuse_a, reuse_b)
  return __builtin_amdgcn_wmma_f32_16x16x4_f32(false, a, false, b, (short)0, c,
                                               false, false);
}

// ---------------- Stage 1: xa[br][m][r] = x @ A_br^T (5 branches) -----------
__global__ __launch_bounds__(256) void lora_stage1(
    const float* __restrict__ x, const float* __restrict__ A_sh,
    const float* __restrict__ A_tasks, float* __restrict__ xa) {
  const int lane = threadIdx.x & 31;
  const int wave = threadIdx.x >> 5;
  const int half = lane >> 4;
  const int l16  = lane & 15;
  const int m0   = (blockIdx.x * 8 + wave) * 16;  // 16-row tile per wave
  const int ko   = 2 * half;                      // K slot: lanes 16-31 -> K=2,3

  v8f zero = {};
  v8f acc0 = zero, acc1 = zero, acc2 = zero, acc3 = zero, acc4 = zero;

  const float* xrow = x + (size_t)(m0 + l16) * D_DIM + ko;
  const float* a_s  = A_sh + (size_t)l16 * D_DIM + ko;
  const float* a_t0 = A_tasks + ((size_t)0 * R_DIM + l16) * D_DIM + ko;
  const float* a_t1 = A_tasks + ((size_t)1 * R_DIM + l16) * D_DIM + ko;
  const float* a_t2 = A_tasks + ((size_t)2 * R_DIM + l16) * D_DIM + ko;
  const float* a_t3 = A_tasks + ((size_t)3 * R_DIM + l16) * D_DIM + ko;

  for (int d = 0; d < D_DIM; d += 4) {
    v2f a = *(const v2f*)(xrow + d);  // A frag reused across all 5 branches
    acc0 = wmma_f32(a, *(const v2f*)(a_s  + d), acc0);
    acc1 = wmma_f32(a, *(const v2f*)(a_t0 + d), acc1);
    acc2 = wmma_f32(a, *(const v2f*)(a_t1 + d), acc2);
    acc3 = wmma_f32(a, *(const v2f*)(a_t2 + d), acc3);
    acc4 = wmma_f32(a, *(const v2f*)(a_t3 + d), acc4);
  }

  v8f accs[5] = {acc0, acc1, acc2, acc3, acc4};
#pragma unroll
  for (int br = 0; br < 5; br++) {
    float* o = xa + ((size_t)br * M_TOT + m0) * R_DIM;
#pragma unroll
    for (int v = 0; v < 8; v++)
      o[(size_t)(v + 8 * half) * R_DIM + l16] = accs[br][v];  // RT: re-read soon
  }
}

// ---------------- Stage 2: main GEMM + bias + 5-way LoRA expansion ----------
#define MBLK 64
#define NBLK 128
#define KT   16
#define LSTR 20  // LDS row pitch in floats: 80B (16B aligned, conflict-free)

__global__ __launch_bounds__(256) void mtlora_main(
    const float* __restrict__ x, const float* __restrict__ W,
    const float* __restrict__ bias, const float* __restrict__ B_sh,
    const float* __restrict__ B_tasks, const float* __restrict__ scales,
    const float* __restrict__ xa, float* __restrict__ out) {
  __shared__ float xs[MBLK * LSTR];   // 64 x KT tile of x
  __shared__ float wsh[NBLK * LSTR];  // 128 x KT tile of W

  const int tid  = threadIdx.x;
  const int lane = tid & 31;
  const int wave = tid >> 5;
  const int half = lane >> 4;
  const int l16  = lane & 15;
  // N is the FAST grid dimension: the 8 blocks sharing one x row-tile launch
  // adjacently so x stays L2-resident across its N sweep (W is L2-resident
  // globally at 4 MB).
  const int blockN = blockIdx.x * NBLK;
  const int blockM = blockIdx.y * MBLK;
  const int mi  = wave >> 1;         // wave's 16-row tile (0..3)
  const int njb = (wave & 1) * 4;    // wave's first 16-col tile (0 or 4)
  const int m0  = blockM + mi * 16;
  const int ko  = 2 * half;

  // Cooperative loader mapping: 4 threads x float4 cover one 16-float row.
  const int lrow = tid >> 2;         // 0..63
  const int lcol = (tid & 3) * 4;    // 0,4,8,12
  const float* xg  = x + (size_t)(blockM + lrow) * D_DIM + lcol;
  const float* wg0 = W + (size_t)(blockN + lrow) * D_DIM + lcol;
  const float* wg1 = W + (size_t)(blockN + 64 + lrow) * D_DIM + lcol;
  float* xs_p  = xs  + lrow * LSTR + lcol;
  float* ws_p0 = wsh + lrow * LSTR + lcol;
  float* ws_p1 = wsh + (64 + lrow) * LSTR + lcol;

  v8f zero = {};
  v8f c0 = zero, c1 = zero, c2 = zero, c3 = zero;

  for (int kt = 0; kt < D_DIM; kt += KT) {
    *(v4f*)xs_p  = *(const v4f*)(xg + kt);
    *(v4f*)ws_p0 = *(const v4f*)(wg0 + kt);
    *(v4f*)ws_p1 = *(const v4f*)(wg1 + kt);
    if (kt + KT < D_DIM) {  // prefetch next chunk behind the WMMA chain
      __builtin_prefetch(xg + kt + KT, 0, 0);
      __builtin_prefetch(wg0 + kt + KT, 0, 0);
      __builtin_prefetch(wg1 + kt + KT, 0, 0);
    }
    __syncthreads();
    const float* xrow = xs + (mi * 16 + l16) * LSTR + ko;
    const float* wr0 = wsh + ((njb + 0) * 16 + l16) * LSTR + ko;
    const float* wr1 = wsh + ((njb + 1) * 16 + l16) * LSTR + ko;
    const float* wr2 = wsh + ((njb + 2) * 16 + l16) * LSTR + ko;
    const float* wr3 = wsh + ((njb + 3) * 16 + l16) * LSTR + ko;
#pragma unroll
    for (int kk = 0; kk < KT; kk += 4) {
      v2f a = *(const v2f*)(xrow + kk);
      c0 = wmma_f32(a, *(const v2f*)(wr0 + kk), c0);
      c1 = wmma_f32(a, *(const v2f*)(wr1 + kk), c1);
      c2 = wmma_f32(a, *(const v2f*)(wr2 + kk), c2);
      c3 = wmma_f32(a, *(const v2f*)(wr3 + kk), c3);
    }
    __syncthreads();
  }

  // Preload this wave's rank-16 A-fragments for all 5 LoRA branches.
  v2f afrag[5][4];
#pragma unroll
  for (int br = 0; br < 5; br++) {
    const float* xr = xa + ((size_t)br * M_TOT + m0 + l16) * R_DIM + ko;
#pragma unroll
    for (int kb = 0; kb < 4; kb++) afrag[br][kb] = *(const v2f*)(xr + kb * 4);
  }
  float sc[4];
#pragma unroll
  for (int t = 0; t < 4; t++) sc[t] = scales[t];

  const size_t MO = (size_t)M_TOT * O_DIM;
  v8f cc[4] = {c0, c1, c2, c3};
#pragma unroll
  for (int j = 0; j < 4; j++) {
    const int n0 = blockN + (njb + j) * 16;
    const float bv = bias[n0 + l16];
    v8f cmain = cc[j];
#pragma unroll
    for (int v = 0; v < 8; v++) cmain[v] += bv;

    // Shared-LoRA branch -> out[0]
    {
      const float* brp = B_sh + (size_t)(n0 + l16) * R_DIM + ko;
      v8f d = zero;
#pragma unroll
      for (int kb = 0; kb < 4; kb++)
        d = wmma_f32(afrag[0][kb], *(const v2f*)(brp + kb * 4), d);
      float* o = out + (size_t)(m0 + 8 * half) * O_DIM + n0 + l16;
#pragma unroll
      for (int v = 0; v < 8; v++)
        __builtin_nontemporal_store(cmain[v] + d[v], o + (size_t)v * O_DIM);
    }
    // Per-task branches -> out[1+t]
#pragma unroll
    for (int t = 0; t < 4; t++) {
      const float* brp = B_tasks + ((size_t)t * O_DIM + n0 + l16) * R_DIM + ko;
      v8f d = zero;
#pragma unroll
      for (int kb = 0; kb < 4; kb++)
        d = wmma_f32(afrag[1 + t][kb], *(const v2f*)(brp + kb * 4), d);
      float* o = out + (size_t)(1 + t) * MO +
                 (size_t)(m0 + 8 * half) * O_DIM + n0 + l16;
#pragma unroll
      for (int v = 0; v < 8; v++)
        __builtin_nontemporal_store(cmain[v] + sc[t] * d[v],
                                    o + (size_t)v * O_DIM);
    }
  }
}

extern "C" void kernel_launch(void* const* d_in, const int* in_sizes, int n_in,
                              void* d_out, int out_size, void* d_ws,
                              size_t ws_size, hipStream_t stream) {
  (void)in_sizes; (void)n_in; (void)out_size; (void)ws_size;
  const float* x        = (const float*)d_in[0];
  const float* W        = (const float*)d_in[1];
  const float* bias     = (const float*)d_in[2];
  const float* A_sh     = (const float*)d_in[3];
  const float* B_sh     = (const float*)d_in[4];
  const float* A_tasks  = (const float*)d_in[5];
  const float* B_tasks  = (const float*)d_in[6];
  const float* scales   = (const float*)d_in[7];
  float* out = (float*)d_out;
  float* xa  = (float*)d_ws;  // 5 * 32768 * 16 floats = 10 MB

  // Stage 1: 2048 M-tiles, 8 waves/block -> 256 blocks.
  lora_stage1<<<dim3(M_TOT / 128), 256, 0, stream>>>(x, A_sh, A_tasks, xa);
  // Stage 2: N fast (8), M slow (512) for L2-friendly x reuse.
  mtlora_main<<<dim3(O_DIM / NBLK, M_TOT / MBLK), 256, 0, stream>>>(
      x, W, bias, B_sh, B_tasks, scales, xa, out);
}